// GSEncoder_75831942578323
// MI455X (gfx1250) — compile-verified
//
#include <hip/hip_runtime.h>
#include <cmath>

#define LVLS 16

typedef float v2f __attribute__((ext_vector_type(2)));
typedef float v4f __attribute__((ext_vector_type(4)));

struct HashParams {
  float    scale[LVLS];
  int      off[LVLS];
  int      stride[LVLS];
  unsigned hashedMask;
};

// One thread per point. 16 levels fully unrolled -> up to 128 independent
// 8-byte gathers in flight per thread (L2-resident table). Streamed outputs
// use non-temporal (TH=NT) stores so the 176MB output stream does not evict
// the 49MB embedding table from L2.
__global__ __launch_bounds__(256) void hashgrid_encode_kernel(
    const float* __restrict__ coords,
    const float* __restrict__ pose,
    const float* __restrict__ emb,
    float* __restrict__ out_enc,
    float* __restrict__ out_pose,
    int N, HashParams P)
{
  const int n = blockIdx.x * blockDim.x + threadIdx.x;
  if (n >= N) return;

  // coords are AoS [N,3]; streamed once -> non-temporal loads
  const float bx = __builtin_nontemporal_load(coords + 3*(size_t)n + 0);
  const float by = __builtin_nontemporal_load(coords + 3*(size_t)n + 1);
  const float bz = __builtin_nontemporal_load(coords + 3*(size_t)n + 2);
  const float x = (bx + 256.0f) * (1.0f / 512.0f);
  const float y = (by + 256.0f) * (1.0f / 512.0f);
  const float z = (bz + 256.0f) * (1.0f / 512.0f);

#pragma unroll
  for (int l = 0; l < LVLS; ++l) {
    const float s  = P.scale[l];
    const float px = x * s + 0.5f;
    const float py = y * s + 0.5f;
    const float pz = z * s + 0.5f;
    const float gx = floorf(px), gy = floorf(py), gz = floorf(pz);
    const float fx = px - gx,    fy = py - gy,    fz = pz - gz;
    const unsigned ix = (unsigned)gx, iy = (unsigned)gy, iz = (unsigned)gz;

    unsigned idx[8];
    if ((P.hashedMask >> l) & 1u) {
      // hashed level: size is exactly 2^19 for this config -> AND mask
      const unsigned hx0 = ix;                      // prime0 == 1
      const unsigned hx1 = ix + 1u;
      const unsigned hy0 = iy * 2654435761u;
      const unsigned hy1 = (iy + 1u) * 2654435761u;
      const unsigned hz0 = iz * 805459861u;
      const unsigned hz1 = (iz + 1u) * 805459861u;
#pragma unroll
      for (int k = 0; k < 8; ++k) {
        const unsigned h = ((k & 4) ? hx1 : hx0) ^
                           ((k & 2) ? hy1 : hy0) ^
                           ((k & 1) ? hz1 : hz0);
        idx[k] = h & 0x7FFFFu;
      }
    } else {
      // dense level: index always < table size -> no modulo needed
      const unsigned st  = (unsigned)P.stride[l];
      const unsigned st2 = st * st;
      const unsigned b   = ix + iy * st + iz * st2;
#pragma unroll
      for (int k = 0; k < 8; ++k) {
        idx[k] = b + ((k >> 2) & 1u)
                   + (((k >> 1) & 1u) ? st  : 0u)
                   + (((k      ) & 1u) ? st2 : 0u);
      }
    }

    const float* base = emb + 2 * (size_t)P.off[l];
    float a0 = 0.0f, a1 = 0.0f;
#pragma unroll
    for (int k = 0; k < 8; ++k) {
      // 8-byte aligned gather (regular temporal load: keep table in L2/WGP$)
      const v2f f = *reinterpret_cast<const v2f*>(base + 2 * (size_t)idx[k]);
      const float wx = (k & 4) ? fx : 1.0f - fx;
      const float wy = (k & 2) ? fy : 1.0f - fy;
      const float wz = (k & 1) ? fz : 1.0f - fz;
      const float w = wx * wy * wz;
      a0 = fmaf(w, f.x, a0);
      a1 = fmaf(w, f.y, a1);
    }

    // store this level's 2 features immediately (keeps VGPR pressure low)
    v2f r; r.x = a0; r.y = a1;
    __builtin_nontemporal_store(
        r, reinterpret_cast<v2f*>(out_enc + (size_t)n * 32 + 2 * l));
  }

  // pose replication: uniform scalar loads, 3 x 128-bit NT stores per point
  v4f p0, p1, p2;
  p0.x = pose[0];  p0.y = pose[1];  p0.z = pose[2];  p0.w = pose[3];
  p1.x = pose[4];  p1.y = pose[5];  p1.z = pose[6];  p1.w = pose[7];
  p2.x = pose[8];  p2.y = pose[9];  p2.z = pose[10]; p2.w = pose[11];
  v4f* dp = reinterpret_cast<v4f*>(out_pose + (size_t)n * 12);
  __builtin_nontemporal_store(p0, dp + 0);
  __builtin_nontemporal_store(p1, dp + 1);
  __builtin_nontemporal_store(p2, dp + 2);
}

extern "C" void kernel_launch(void* const* d_in, const int* in_sizes, int n_in,
                              void* d_out, int out_size, void* d_ws, size_t ws_size,
                              hipStream_t stream) {
  const float* coords = (const float*)d_in[0];
  const float* pose   = (const float*)d_in[1];
  const float* emb    = (const float*)d_in[2];
  float* out = (float*)d_out;
  const int N = in_sizes[0] / 3;

  // Mirror the reference's double-precision level-config computation exactly.
  HashParams P{};
  const double PS = std::exp2(std::log2(2048.0 / 16.0) / 15.0);
  long long off = 0;
  unsigned hm = 0;
  for (int l = 0; l < LVLS; ++l) {
    const double s = 16.0 * std::pow(PS, (double)l) - 1.0;
    const int r = (int)std::ceil(s) + 1;
    const long long p3 = (long long)(r + 1) * (r + 1) * (r + 1);
    long long p = (p3 <= 524288LL) ? p3 : 524288LL;
    if (p3 > 524288LL) hm |= (1u << l);
    p = (p + 7) / 8 * 8;  // align to 8 like torch-ngp
    P.scale[l]  = (float)s;
    P.stride[l] = r + 1;
    P.off[l]    = (int)off;
    off += p;
  }
  P.hashedMask = hm;

  float* out_enc  = out;
  float* out_pose = out + (size_t)N * 32;

  dim3 block(256);
  dim3 grid((N + 255) / 256);
  hipLaunchKernelGGL(hashgrid_encode_kernel, grid, block, 0, stream,
                     coords, pose, emb, out_enc, out_pose, N, P);
}